// SimpleScheduler_L_34213709480230
// MI455X (gfx1250) — compile-verified
//
#include <hip/hip_runtime.h>

// Problem constants from the reference.
#define BATCH   16384
#define DIM     4096
#define SLAY    16          // S = H - P
#define HLAY    32
#define KSPLIT  4
#define KCHUNK  (DIM / KSPLIT)   // 1024
#define NEGV    (-1e9f)
#define TAUV    5.0f

typedef __attribute__((ext_vector_type(2))) float v2f;
typedef __attribute__((ext_vector_type(8))) float v8f;

// ---------------------------------------------------------------------------
// Kernel A: logits partial GEMM via V_WMMA_F32_16X16X4_F32 (fp32-exact path).
//   part[KSPLIT][BATCH][SLAY] partial sums in workspace.
// One wave handles one (16-row tile, K-slice). 8 waves / block.
// ---------------------------------------------------------------------------
__global__ __launch_bounds__(256) void gemm_logits_kernel(
    const float* __restrict__ x,      // [BATCH, DIM]
    const float* __restrict__ W,      // [DIM, SLAY]
    float* __restrict__ part)         // [KSPLIT, BATCH, SLAY]
{
    const int wid  = blockIdx.x * (blockDim.x >> 5) + (threadIdx.x >> 5);
    const int tile = wid >> 2;            // 0 .. BATCH/16 - 1
    const int ks   = wid & (KSPLIT - 1);  // K slice
    const int lane = threadIdx.x & 31;
    const int half = lane >> 4;           // 0 or 1 (selects K pair)
    const int sub  = lane & 15;           // M row (A) / N col (B)

    const int m0 = tile * 16;
    const int k0 = ks * KCHUNK;

    // A layout (32-bit, 16x4): lane L holds M=L&15; VGPR0/1 = K = (L>>4)*2 + {0,1}
    const float* aptr = x + (size_t)(m0 + sub) * DIM + k0 + (half << 1);
    // B layout (4x16): lane L holds N=L&15; VGPR0/1 = K = (L>>4)*2 + {0,1}
    const float* bptr = W + (size_t)(half << 1) * SLAY + sub;

    v8f acc = {};
#pragma unroll 8
    for (int k = 0; k < KCHUNK; k += 4) {
        __builtin_prefetch(aptr + k + 512, 0, 1);   // global_prefetch_b8
        v2f a = *(const v2f*)(aptr + k);            // K, K+1 for this lane
        v2f b;
        b.x = bptr[(size_t)k * SLAY];               // W[k+2*half    , sub]
        b.y = bptr[(size_t)k * SLAY + SLAY];        // W[k+2*half + 1, sub]
        acc = __builtin_amdgcn_wmma_f32_16x16x4_f32(
            false, a, false, b, (short)0, acc, false, false);
    }

    // C/D layout: VGPR r -> M = r + 8*half (lanes 0-15 : M=0..7, 16-31 : M=8..15),
    // N = lane & 15.
    float* op = part + ((size_t)ks * BATCH + m0 + (half << 3)) * SLAY + sub;
#pragma unroll
    for (int r = 0; r < 8; ++r)
        op[(size_t)r * SLAY] = acc[r];
}

// ---------------------------------------------------------------------------
// Kernel B: per-sample Gumbel top-n-without-replacement + broadcast to output.
// One wave per sample; lanes 0-15 carry the 16 sub-layer slots (upper half
// mirrors). Output row = [HLAY, 2, NH] = 2048 floats, written as b128 stores.
// ---------------------------------------------------------------------------
__global__ __launch_bounds__(256) void sample_out_kernel(
    const float* __restrict__ part,     // [KSPLIT, BATCH, SLAY]
    const float* __restrict__ latency,  // [BATCH]
    const float* __restrict__ b_head,   // [SLAY]
    const float* __restrict__ gumbel,   // [SLAY, BATCH, SLAY]
    float* __restrict__ out)            // [BATCH, HLAY, 2, 32]
{
    const int b    = blockIdx.x * (blockDim.x >> 5) + (threadIdx.x >> 5);
    const int lane = threadIdx.x & 31;
    const int s    = lane & 15;

    // Deterministic fixed-order split-K reduction + bias.
    float logit = b_head[s];
#pragma unroll
    for (int j = 0; j < KSPLIT; ++j)
        logit += part[((size_t)j * BATCH + b) * SLAY + s];

    // n = clip(round(latency*H), P, H) - P   (round-half-even == rintf)
    const float nf = fminf(fmaxf(rintf(latency[b] * 32.0f), 16.0f), 32.0f) - 16.0f;

    float mask = 0.0f;
    for (int t = 0; t < SLAY; ++t) {
        const float g = gumbel[((size_t)t * BATCH + b) * SLAY + s];
        const float z = (logit + mask * NEGV + g) / TAUV;

        // max over the 16 slots (xor masks 8..1 stay inside each half-wave)
        float m = z;
#pragma unroll
        for (int o = 8; o >= 1; o >>= 1) m = fmaxf(m, __shfl_xor(m, o, 32));
        const float e = __expf(z - m) == 0.0f ? expf(z - m) : expf(z - m);
        float sum = e;
#pragma unroll
        for (int o = 8; o >= 1; o >>= 1) sum += __shfl_xor(sum, o, 32);
        const float soft = e / sum;

        float ms = soft;
#pragma unroll
        for (int o = 8; o >= 1; o >>= 1) ms = fmaxf(ms, __shfl_xor(ms, o, 32));
        // first index attaining the max (matches jnp.argmax tie rule)
        const unsigned long long bal = __ballot(soft == ms);
        const int idx = __ffsll(bal & 0xFFFFull) - 1;

        const float h = (s == idx) ? 1.0f : 0.0f;
        const float sample = soft + (h - soft);   // straight-through, exact order
        if ((float)t < nf) mask += sample;
    }

    // Broadcast: h < 16 -> 1.0 ; h >= 16 -> mask[h-16], 64 copies each.
    float* row = out + (size_t)b * (HLAY * 2 * 32);
#pragma unroll
    for (int it = 0; it < 16; ++it) {
        const int p  = it * 128 + lane * 4;   // 4-float chunk, h uniform per chunk
        const int hh = p >> 6;
        const float v = (hh < 16) ? 1.0f : __shfl(mask, hh & 15, 32);
        float4 vv = make_float4(v, v, v, v);
        *(float4*)(row + p) = vv;
    }
}

// ---------------------------------------------------------------------------
extern "C" void kernel_launch(void* const* d_in, const int* in_sizes, int n_in,
                              void* d_out, int out_size, void* d_ws, size_t ws_size,
                              hipStream_t stream) {
    const float* x       = (const float*)d_in[0];
    const float* latency = (const float*)d_in[1];
    const float* W_head  = (const float*)d_in[2];
    const float* b_head  = (const float*)d_in[3];
    const float* gumbel  = (const float*)d_in[4];
    float* out = (float*)d_out;
    float* part = (float*)d_ws;   // KSPLIT * BATCH * SLAY floats = 4 MB

    // 1024 tiles * KSPLIT waves, 8 waves per 256-thread block.
    const int gemm_waves  = (BATCH / 16) * KSPLIT;
    gemm_logits_kernel<<<gemm_waves / 8, 256, 0, stream>>>(x, W_head, part);

    // One wave per sample.
    sample_out_kernel<<<BATCH / 8, 256, 0, stream>>>(part, latency, b_head,
                                                     gumbel, out);
}